// GIN_10170482557046
// MI455X (gfx1250) — compile-verified
//
#include <hip/hip_runtime.h>
#include <hip/hip_bf16.h>

// ---------------------------------------------------------------------------
// GIN forward on MI455X (gfx1250, wave32).
//   scatter-add (memory bound, ~820MB -> ~35us @ 23.3TB/s)
//   node MLPs via V_WMMA_F32_16X16X4_F32 (f32-precision WMMA), fused BN+ReLU
//   mean-pool + hop-weighted heads
// ---------------------------------------------------------------------------

#define N_NODES 50000
#define N_EDGES 800000
#define FDIM    64
#define HDIM    64
#define CDIM    10
#define BGRAPH  512
#define BN_EPS  1e-5f

typedef __attribute__((ext_vector_type(2))) float v2f;
typedef __attribute__((ext_vector_type(8))) float v8f;

// ---------------------------------------------------------------------------
// Edge scatter-add: aggr[dst] += h[src].  1 thread per (edge, 4-feature chunk)
// ---------------------------------------------------------------------------
__global__ __launch_bounds__(256) void gin_scatter(
    const float* __restrict__ hin, const int* __restrict__ src,
    const int* __restrict__ dst, float* __restrict__ aggr)
{
    int t = blockIdx.x * 256 + threadIdx.x;        // 12.8M threads total
    if (t >= N_EDGES * 16) return;
    int e  = t >> 4;
    int c4 = (t & 15) << 2;
    int s = src[e];
    int d = dst[e];
    const float4 v = *(const float4*)(hin + (size_t)s * HDIM + c4);
    float* o = aggr + (size_t)d * HDIM + c4;
    atomicAdd(o + 0, v.x);
    atomicAdd(o + 1, v.y);
    atomicAdd(o + 2, v.z);
    atomicAdd(o + 3, v.w);
}

// ---------------------------------------------------------------------------
// One wave per 16-node tile:
//   z  = hin + aggr                             (16x64, staged in LDS)
//   t  = relu(bn1(z @ w1 + b1))                 (WMMA f32 16x16x4, K=64)
//   ho = relu(bn2(t @ w2 + b2))                 (WMMA f32 16x16x4, K=64)
// Fragment layouts per CDNA5 ISA 7.12.2:
//   A 16x4 : lanes 0-15 -> M=lane, {K0,K1}; lanes 16-31 -> M=lane-16, {K2,K3}
//   B 4x16 : lanes 0-15 -> N=lane, {K0,K1}; lanes 16-31 -> N=lane-16, {K2,K3}
//   C/D    : vgpr i -> M = i + 8*(lane/16), N = lane%16  (column fixed / lane)
// ---------------------------------------------------------------------------
__global__ __launch_bounds__(32) void gin_mlp_wmma(
    const float* __restrict__ hin,  const float* __restrict__ aggr,
    const float* __restrict__ w1,   const float* __restrict__ b1,
    const float* __restrict__ bng,  const float* __restrict__ bnb,
    const float* __restrict__ rm,   const float* __restrict__ rv,
    const float* __restrict__ w2,   const float* __restrict__ b2,
    const float* __restrict__ og,   const float* __restrict__ ob,
    const float* __restrict__ orm,  const float* __restrict__ orv,
    float* __restrict__ hout)
{
    __shared__ float zt[16][HDIM + 4];   // +4 pad: rotate LDS banks per row
    __shared__ float tt[16][HDIM + 4];

    const int lane = threadIdx.x;        // 0..31, no divergence anywhere
    const int half = lane >> 4;          // 0 / 1
    const int l16  = lane & 15;
    const int row0 = blockIdx.x * 16;    // N_NODES == 3125 * 16 exactly

    // Warm the WGP caches with the two 16KB weight matrices.
    __builtin_prefetch(w1 + lane * 32, 0, 1);
    __builtin_prefetch(w2 + lane * 32, 0, 1);

    // Stage z = hin + aggr (16 rows x 64 cols) into LDS, float4-vectorized.
    for (int i = lane; i < 16 * 16; i += 32) {
        int r  = i >> 4;
        int c4 = (i & 15) << 2;
        const float4 a = *(const float4*)(hin  + (size_t)(row0 + r) * HDIM + c4);
        const float4 g = *(const float4*)(aggr + (size_t)(row0 + r) * HDIM + c4);
        zt[r][c4 + 0] = a.x + g.x;
        zt[r][c4 + 1] = a.y + g.y;
        zt[r][c4 + 2] = a.z + g.z;
        zt[r][c4 + 3] = a.w + g.w;
    }
    __syncthreads();

    // GEMM 1: tt = relu(bn1(zt @ w1 + b1))
    for (int nt = 0; nt < 4; ++nt) {
        const int col = nt * 16 + l16;
        v8f acc = {};
        #pragma unroll
        for (int kt = 0; kt < 16; ++kt) {
            const int k0 = kt * 4 + half * 2;
            v2f a, b;
            a.x = zt[l16][k0];
            a.y = zt[l16][k0 + 1];
            b.x = w1[(k0    ) * HDIM + col];
            b.y = w1[(k0 + 1) * HDIM + col];
            acc = __builtin_amdgcn_wmma_f32_16x16x4_f32(
                false, a, false, b, (short)0, acc, false, false);
        }
        // Fused bias + eval-mode BN + ReLU (column constant per lane).
        const float sc = bng[col] * rsqrtf(rv[col] + BN_EPS);
        const float sh = bnb[col] + (b1[col] - rm[col]) * sc;
        #pragma unroll
        for (int i = 0; i < 8; ++i) {
            const int m = i + half * 8;
            tt[m][col] = fmaxf(acc[i] * sc + sh, 0.0f);
        }
    }
    __syncthreads();

    // GEMM 2: hout = relu(bn2(tt @ w2 + b2))
    for (int nt = 0; nt < 4; ++nt) {
        const int col = nt * 16 + l16;
        v8f acc = {};
        #pragma unroll
        for (int kt = 0; kt < 16; ++kt) {
            const int k0 = kt * 4 + half * 2;
            v2f a, b;
            a.x = tt[l16][k0];
            a.y = tt[l16][k0 + 1];
            b.x = w2[(k0    ) * HDIM + col];
            b.y = w2[(k0 + 1) * HDIM + col];
            acc = __builtin_amdgcn_wmma_f32_16x16x4_f32(
                false, a, false, b, (short)0, acc, false, false);
        }
        const float sc = og[col] * rsqrtf(orv[col] + BN_EPS);
        const float sh = ob[col] + (b2[col] - orm[col]) * sc;
        #pragma unroll
        for (int i = 0; i < 8; ++i) {
            const int m = i + half * 8;
            hout[(size_t)(row0 + m) * HDIM + col] = fmaxf(acc[i] * sc + sh, 0.0f);
        }
    }
}

// ---------------------------------------------------------------------------
// Per-graph node counts (clamped to >= 1 at use site).
// ---------------------------------------------------------------------------
__global__ __launch_bounds__(256) void gin_counts(
    const int* __restrict__ batch, float* __restrict__ counts)
{
    int n = blockIdx.x * 256 + threadIdx.x;
    if (n < N_NODES) atomicAdd(&counts[batch[n]], 1.0f);
}

// ---------------------------------------------------------------------------
// Segment-sum pooling for x, h1, h2 in one pass (thread per node-feature).
// ---------------------------------------------------------------------------
__global__ __launch_bounds__(256) void gin_pool(
    const float* __restrict__ x,  const float* __restrict__ h1,
    const float* __restrict__ h2, const int* __restrict__ batch,
    float* __restrict__ p0, float* __restrict__ p1, float* __restrict__ p2)
{
    int t = blockIdx.x * 256 + threadIdx.x;
    if (t >= N_NODES * HDIM) return;
    int n = t >> 6;
    int f = t & 63;
    int g = batch[n];
    atomicAdd(&p0[g * HDIM + f], x[t]);
    atomicAdd(&p1[g * HDIM + f], h1[t]);
    atomicAdd(&p2[g * HDIM + f], h2[t]);
}

// ---------------------------------------------------------------------------
// Heads: out[g,c] = sum_i hopw[g,i] * (pooled_i[g]/cnt[g] @ li_w + li_b)[c]
// ---------------------------------------------------------------------------
__global__ __launch_bounds__(32) void gin_head(
    const float* __restrict__ p0, const float* __restrict__ p1,
    const float* __restrict__ p2, const float* __restrict__ counts,
    const float* __restrict__ hopw,
    const float* __restrict__ l0w, const float* __restrict__ l0b,
    const float* __restrict__ l1w, const float* __restrict__ l1b,
    const float* __restrict__ l2w, const float* __restrict__ l2b,
    float* __restrict__ out)
{
    const int g = blockIdx.x;
    const int c = threadIdx.x;
    if (c >= CDIM) return;
    const float inv = 1.0f / fmaxf(counts[g], 1.0f);
    float s0 = l0b[c], s1 = l1b[c], s2 = l2b[c];
    for (int f = 0; f < HDIM; ++f) {
        s0 = fmaf(p0[g * HDIM + f] * inv, l0w[f * CDIM + c], s0);
        s1 = fmaf(p1[g * HDIM + f] * inv, l1w[f * CDIM + c], s1);
        s2 = fmaf(p2[g * HDIM + f] * inv, l2w[f * CDIM + c], s2);
    }
    out[g * CDIM + c] = s0 * hopw[g * 3 + 0]
                      + s1 * hopw[g * 3 + 1]
                      + s2 * hopw[g * 3 + 2];
}

// ---------------------------------------------------------------------------
extern "C" void kernel_launch(void* const* d_in, const int* in_sizes, int n_in,
                              void* d_out, int out_size, void* d_ws, size_t ws_size,
                              hipStream_t stream)
{
    // Inputs in setup_inputs() dict order:
    const float* x     = (const float*)d_in[0];
    const int*   ei    = (const int*)  d_in[1];   // 2 x E, row-major
    const int*   batch = (const int*)  d_in[2];
    const float* hopw  = (const float*)d_in[3];
    // conv0: 4..15, conv1: 16..27
    const float *c0_w1 = (const float*)d_in[4],  *c0_b1 = (const float*)d_in[5];
    const float *c0_g  = (const float*)d_in[6],  *c0_b  = (const float*)d_in[7];
    const float *c0_rm = (const float*)d_in[8],  *c0_rv = (const float*)d_in[9];
    const float *c0_w2 = (const float*)d_in[10], *c0_b2 = (const float*)d_in[11];
    const float *o0_g  = (const float*)d_in[12], *o0_b  = (const float*)d_in[13];
    const float *o0_rm = (const float*)d_in[14], *o0_rv = (const float*)d_in[15];
    const float *c1_w1 = (const float*)d_in[16], *c1_b1 = (const float*)d_in[17];
    const float *c1_g  = (const float*)d_in[18], *c1_b  = (const float*)d_in[19];
    const float *c1_rm = (const float*)d_in[20], *c1_rv = (const float*)d_in[21];
    const float *c1_w2 = (const float*)d_in[22], *c1_b2 = (const float*)d_in[23];
    const float *o1_g  = (const float*)d_in[24], *o1_b  = (const float*)d_in[25];
    const float *o1_rm = (const float*)d_in[26], *o1_rv = (const float*)d_in[27];
    const float *l0w = (const float*)d_in[28], *l0b = (const float*)d_in[29];
    const float *l1w = (const float*)d_in[30], *l1b = (const float*)d_in[31];
    const float *l2w = (const float*)d_in[32], *l2b = (const float*)d_in[33];

    const int* src = ei;
    const int* dst = ei + N_EDGES;

    // Workspace layout (floats).
    float* ws     = (float*)d_ws;
    float* aggr   = ws;                                  // N*H
    float* h1     = aggr + (size_t)N_NODES * HDIM;       // N*H
    float* h2     = h1   + (size_t)N_NODES * HDIM;       // N*H
    float* p0     = h2   + (size_t)N_NODES * HDIM;       // B*H
    float* p1     = p0   + (size_t)BGRAPH * HDIM;
    float* p2     = p1   + (size_t)BGRAPH * HDIM;
    float* counts = p2   + (size_t)BGRAPH * HDIM;        // B

    const int scat_grid = (N_EDGES * 16 + 255) / 256;    // 50000 blocks
    const int mlp_grid  = N_NODES / 16;                  // 3125 tiles exactly

    // ---- Layer 0 ----
    hipMemsetAsync(aggr, 0, (size_t)N_NODES * HDIM * sizeof(float), stream);
    gin_scatter<<<scat_grid, 256, 0, stream>>>(x, src, dst, aggr);
    gin_mlp_wmma<<<mlp_grid, 32, 0, stream>>>(
        x, aggr, c0_w1, c0_b1, c0_g, c0_b, c0_rm, c0_rv,
        c0_w2, c0_b2, o0_g, o0_b, o0_rm, o0_rv, h1);

    // ---- Layer 1 ----
    hipMemsetAsync(aggr, 0, (size_t)N_NODES * HDIM * sizeof(float), stream);
    gin_scatter<<<scat_grid, 256, 0, stream>>>(h1, src, dst, aggr);
    gin_mlp_wmma<<<mlp_grid, 32, 0, stream>>>(
        h1, aggr, c1_w1, c1_b1, c1_g, c1_b, c1_rm, c1_rv,
        c1_w2, c1_b2, o1_g, o1_b, o1_rm, o1_rv, h2);

    // ---- Pooling + heads ----
    hipMemsetAsync(p0, 0, ((size_t)3 * BGRAPH * HDIM + BGRAPH) * sizeof(float),
                   stream);
    gin_counts<<<(N_NODES + 255) / 256, 256, 0, stream>>>(batch, counts);
    gin_pool<<<(N_NODES * HDIM + 255) / 256, 256, 0, stream>>>(
        x, h1, h2, batch, p0, p1, p2);
    gin_head<<<BGRAPH, 32, 0, stream>>>(
        p0, p1, p2, counts, hopw, l0w, l0b, l1w, l1b, l2w, l2b, (float*)d_out);
}